// ResidualVectorQuantizer_7902739824872
// MI455X (gfx1250) — compile-verified
//
#include <hip/hip_runtime.h>
#include <hip/hip_bf16.h>

// Problem constants (from reference setup_inputs)
constexpr int B_ = 16, T_ = 2048, D_ = 128, Q_ = 8, K_ = 1024;
constexpr int N_ = B_ * T_;            // 32768 tokens
constexpr float DECAY_ = 0.99f;
constexpr float OMD_   = 0.01f;        // 1 - decay
constexpr float EPS_   = 1e-10f;

typedef __attribute__((ext_vector_type(2))) float v2f;
typedef __attribute__((ext_vector_type(4))) float v4f;
typedef __attribute__((ext_vector_type(8))) float v8f;

// ---------------------------------------------------------------- zero fill
__global__ void rvq_zero(float* __restrict__ p, int n) {
    int i = blockIdx.x * blockDim.x + threadIdx.x;
    if (i < n) p[i] = 0.0f;
}

// ------------------------------------------------- ||w||^2 per code, all Q
__global__ void rvq_wnorm(const float* __restrict__ w, float* __restrict__ wn) {
    int idx = blockIdx.x * blockDim.x + threadIdx.x;   // 0 .. Q*K
    if (idx >= Q_ * K_) return;
    const float* row = w + (size_t)idx * D_;
    float s = 0.0f;
#pragma unroll
    for (int d = 0; d < D_; d += 4) {
        v4f v = *(const v4f*)(row + d);
        s += v.x * v.x + v.y * v.y + v.z * v.z + v.w * v.w;
    }
    wn[idx] = s;
}

// ---------------------------------------------------------------------------
// Distance + argmin via V_WMMA_F32_16X16X4_F32.
//
// One wave = one 16-token tile (A in 64 VGPRs). The 16-code weight panel
// (8 KB) is shared by all 8 waves of the block: DMA'd into LDS with CDNA5
// async global->LDS loads (ASYNCcnt), double-buffered, stored in WMMA
// *fragment order* ([s][lane], 8B cells) so ds_load_b64 is bank-conflict
// free. The next panel's DMA overlaps the current 32-WMMA chain, and the
// fragment reads are software-pipelined (4-fragment groups, rotated) so the
// DS latency is covered by WMMA execution instead of s_wait_dscnt 0x0.
// ---------------------------------------------------------------------------
__global__ __launch_bounds__(256) void rvq_dist_argmin(
    const float* __restrict__ r, const float* __restrict__ w,
    const float* __restrict__ wn, int* __restrict__ codes)
{
    __shared__ v2f panel[2][32 * 32];   // [buf][s*32 + lane], 8 KB per buffer

    const int lane = threadIdx.x & 31;
    const int wave = threadIdx.x >> 5;
    const int base = (blockIdx.x * 8 + wave) * 16;     // token tile base
    const int row  = lane & 15;
    const int hi   = lane >> 4;

    // A fragments: lane holds r[base+row][4s + 2*hi .. +1]  (ISA 16x4 A layout)
    const float* rrow = r + (size_t)(base + row) * D_ + 2 * hi;
    v2f a[32];
#pragma unroll
    for (int s = 0; s < 32; ++s) a[s] = *(const v2f*)(rrow + 4 * s);

    // Stage column-tile `ct` of w into panel[buf] in fragment order.
    // Wave `wave` stages s = wave*4 .. wave*4+3 (4 async b64 ops, 32 lanes each).
    auto stage = [&](int ct, int buf) {
        const float* gbase = w + (size_t)(ct * 16 + row) * D_ + 2 * hi;
#pragma unroll
        for (int t = 0; t < 4; ++t) {
            const int s = wave * 4 + t;
            unsigned lds = (unsigned)(uintptr_t)&panel[buf][s * 32 + lane];
            unsigned long long ga = (unsigned long long)(uintptr_t)(gbase + 4 * s);
            asm volatile("global_load_async_to_lds_b64 %0, %1, off"
                         :: "v"(lds), "v"(ga) : "memory");
        }
    };

    stage(0, 0);

    float bestv[8];
    int   bestk[8];
#pragma unroll
    for (int v = 0; v < 8; ++v) { bestv[v] = 3.4e38f; bestk[v] = 0; }

    for (int ct = 0; ct < K_ / 16; ++ct) {
        // panel `ct` DMA complete (issued last iteration), all waves synced.
        asm volatile("s_wait_asynccnt 0x0" ::: "memory");
        __syncthreads();
        // Safe to overwrite buffer (ct+1)&1: last read in iteration ct-1,
        // and every wave passed the barrier above after finishing it.
        if (ct + 1 < K_ / 16) stage(ct + 1, (ct + 1) & 1);

        // Software-pipelined fragment reads: rotate two 4-fragment groups so
        // group g+1's ds_load_b64 issue before group g's WMMAs consume.
        const v2f* bp = &panel[ct & 1][lane];
        v2f bb[2][4];
#pragma unroll
        for (int j = 0; j < 4; ++j) bb[0][j] = bp[j * 32];

        v8f acc = {};
#pragma unroll
        for (int g = 0; g < 8; ++g) {
            const int cur = g & 1;
            if (g + 1 < 8) {
#pragma unroll
                for (int j = 0; j < 4; ++j)
                    bb[cur ^ 1][j] = bp[(g * 4 + 4 + j) * 32];
            }
#pragma unroll
            for (int j = 0; j < 4; ++j)
                acc = __builtin_amdgcn_wmma_f32_16x16x4_f32(
                    false, a[g * 4 + j], false, bb[cur][j], (short)0, acc,
                    false, false);
        }

        const int col = ct * 16 + row;                 // this lane's column
        const float c = wn[col];                       // N = lane&15 for all 8 rows
#pragma unroll
        for (int v = 0; v < 8; ++v) {
            float d = c - 2.0f * acc[v];
            if (d < bestv[v]) { bestv[v] = d; bestk[v] = col; }
        }
    }

    // Argmin reduce across the 16 lanes of each half-wave (xor 1,2,4,8)
#pragma unroll
    for (int v = 0; v < 8; ++v) {
#pragma unroll
        for (int off = 1; off < 16; off <<= 1) {
            float ov = __shfl_xor(bestv[v], off, 32);
            int   ok = __shfl_xor(bestk[v], off, 32);
            if (ov < bestv[v] || (ov == bestv[v] && ok < bestk[v])) {
                bestv[v] = ov; bestk[v] = ok;
            }
        }
    }
    if (row == 0) {                                    // lane 0: rows 0-7, lane 16: rows 8-15
#pragma unroll
        for (int v = 0; v < 8; ++v)
            codes[base + 8 * hi + v] = bestk[v];
    }
}

// ---------------------------------------------------------------------------
// Per-stage stats + residual update: one wave per token.
// Accumulates segment sums (count, sum of r) with L2 float atomics using the
// PRE-update residual, then r -= w[k].
// ---------------------------------------------------------------------------
__global__ __launch_bounds__(256) void rvq_stats_update(
    float* __restrict__ r, const float* __restrict__ w,
    const int* __restrict__ codes,
    float* __restrict__ cc_acc, float* __restrict__ rm_acc)
{
    const int lane = threadIdx.x & 31;
    const int wave = threadIdx.x >> 5;
    const int n = blockIdx.x * 8 + wave;
    const int k = codes[n];

    float* rrow  = r + (size_t)n * D_;
    const float* wrow = w + (size_t)k * D_;
    float* rmrow = rm_acc + (size_t)k * D_;

    const int d = lane * 4;
    v4f rv = *(const v4f*)(rrow + d);
    v4f wv = *(const v4f*)(wrow + d);

    atomicAdd(rmrow + d + 0, rv.x);
    atomicAdd(rmrow + d + 1, rv.y);
    atomicAdd(rmrow + d + 2, rv.z);
    atomicAdd(rmrow + d + 3, rv.w);

    v4f nr = rv - wv;
    *(v4f*)(rrow + d) = nr;

    if (lane == 0) atomicAdd(cc_acc + k, 1.0f);
}

// ------------------------------------------------------------- EMA updates
__global__ void rvq_ema(const float* __restrict__ cc_in,
                        const float* __restrict__ rm_in,
                        const float* __restrict__ cc_acc,
                        const float* __restrict__ rm_acc,
                        float* __restrict__ out_cc,
                        float* __restrict__ out_rm,
                        float* __restrict__ out_w)
{
    int idx = blockIdx.x * blockDim.x + threadIdx.x;   // 0 .. Q*K
    if (idx >= Q_ * K_) return;
    const float invN = 1.0f / (float)N_;
    float ncc = cc_in[idx] * DECAY_ + (cc_acc[idx] * invN) * OMD_;
    out_cc[idx] = ncc;
    float inv = 1.0f / (EPS_ + ncc);
    size_t o = (size_t)idx * D_;
#pragma unroll
    for (int d = 0; d < D_; d += 4) {
        v4f rm = *(const v4f*)(rm_in + o + d);
        v4f ra = *(const v4f*)(rm_acc + o + d);
        v4f nrm = rm * DECAY_ + ra * (OMD_ * invN);
        *(v4f*)(out_rm + o + d) = nrm;
        *(v4f*)(out_w  + o + d) = nrm * inv;
    }
}

// --------------------------------------- quantized = x - r, loss partials
__global__ __launch_bounds__(256) void rvq_finalize(
    const float* __restrict__ x, const float* __restrict__ r,
    float* __restrict__ out_q, float* __restrict__ loss_acc)
{
    int i = (blockIdx.x * blockDim.x + threadIdx.x) * 4;
    v4f rv = *(const v4f*)(r + i);
    v4f xv = *(const v4f*)(x + i);
    v4f q = xv - rv;
    *(v4f*)(out_q + i) = q;
    float s = rv.x * rv.x + rv.y * rv.y + rv.z * rv.z + rv.w * rv.w;
#pragma unroll
    for (int off = 16; off >= 1; off >>= 1) s += __shfl_xor(s, off, 32);
    if ((threadIdx.x & 31) == 0) atomicAdd(loss_acc, s);
}

__global__ void rvq_codes_out(const int* __restrict__ codes, float* __restrict__ out) {
    int i = blockIdx.x * blockDim.x + threadIdx.x;     // 0 .. N*Q
    if (i >= N_ * Q_) return;
    int n = i / Q_, q = i % Q_;
    out[i] = (float)codes[q * N_ + n];                 // [B,T,Q] layout
}

__global__ void rvq_loss_out(const float* __restrict__ loss_acc, float* __restrict__ out) {
    out[0] = loss_acc[0] * (1.0f / ((float)N_ * (float)D_));
}

// ---------------------------------------------------------------------------
extern "C" void kernel_launch(void* const* d_in, const int* in_sizes, int n_in,
                              void* d_out, int out_size, void* d_ws, size_t ws_size,
                              hipStream_t stream) {
    (void)in_sizes; (void)n_in; (void)out_size; (void)ws_size;
    const float* x            = (const float*)d_in[0];
    const float* weight       = (const float*)d_in[1];
    const float* running_mean = (const float*)d_in[2];
    const float* code_count   = (const float*)d_in[3];
    float* out = (float*)d_out;

    // workspace layout (floats): r | rm_acc | cc_acc | loss_acc | wnorm | codes(int)
    float* ws       = (float*)d_ws;
    float* r        = ws;                                  // N*D
    float* rm_acc   = r + (size_t)N_ * D_;                 // Q*K*D
    float* cc_acc   = rm_acc + (size_t)Q_ * K_ * D_;       // Q*K
    float* loss_acc = cc_acc + (size_t)Q_ * K_;            // 1
    float* wnorm    = loss_acc + 1;                        // Q*K
    int*   codes    = (int*)(wnorm + (size_t)Q_ * K_);     // N*Q

    // r = x ; zero accumulators (rm_acc, cc_acc, loss_acc are contiguous)
    hipMemcpyAsync(r, x, (size_t)N_ * D_ * sizeof(float),
                   hipMemcpyDeviceToDevice, stream);
    const int nz = Q_ * K_ * D_ + Q_ * K_ + 1;
    rvq_zero<<<(nz + 255) / 256, 256, 0, stream>>>(rm_acc, nz);

    // ||w||^2 for every code of every stage (weights are fixed within the pass)
    rvq_wnorm<<<(Q_ * K_ + 255) / 256, 256, 0, stream>>>(weight, wnorm);

    // Sequential quantizer stages
    for (int i = 0; i < Q_; ++i) {
        const float* wi = weight + (size_t)i * K_ * D_;
        rvq_dist_argmin<<<N_ / 128, 256, 0, stream>>>(
            r, wi, wnorm + (size_t)i * K_, codes + (size_t)i * N_);
        rvq_stats_update<<<N_ / 8, 256, 0, stream>>>(
            r, wi, codes + (size_t)i * N_,
            cc_acc + (size_t)i * K_, rm_acc + (size_t)i * K_ * D_);
    }

    // Output offsets (flat, in return order)
    const size_t off_codes = (size_t)N_ * D_;
    const size_t off_loss  = off_codes + (size_t)N_ * Q_;
    const size_t off_w     = off_loss + 1;
    const size_t off_rm    = off_w + (size_t)Q_ * K_ * D_;
    const size_t off_cc    = off_rm + (size_t)Q_ * K_ * D_;

    rvq_finalize<<<(N_ * D_) / 1024, 256, 0, stream>>>(x, r, out, loss_acc);
    rvq_ema<<<(Q_ * K_ + 255) / 256, 256, 0, stream>>>(
        code_count, running_mean, cc_acc, rm_acc,
        out + off_cc, out + off_rm, out + off_w);
    rvq_codes_out<<<(N_ * Q_ + 255) / 256, 256, 0, stream>>>(codes, out + off_codes);
    rvq_loss_out<<<1, 1, 0, stream>>>(loss_acc, out + off_loss);
}